// VectorQuantizerEMA_26869315404443
// MI455X (gfx1250) — compile-verified
//
#include <hip/hip_runtime.h>
#include <hip/hip_bf16.h>

typedef float v2f __attribute__((ext_vector_type(2)));
typedef float v8f __attribute__((ext_vector_type(8)));

#define N_ROWS   131072      // 32*64*64
#define DDIM     64
#define KCODES   1024
#define TILES_K  (KCODES / 16)        // 64 column tiles
#define WAVES_PB 8
#define BLOCK    256
#define TILES_PW (TILES_K / WAVES_PB) // 8 tiles per wave
#define NBLOCKS  (N_ROWS / 16)        // 8192

// d_out layout (floats), reference tuple order:
// quantized_st [N_ROWS*DDIM], loss [1], perplexity [1],
// encodings [N_ROWS*KCODES], encoding_indices [N_ROWS]
#define OFF_Q    ((size_t)0)
#define OFF_LOSS ((size_t)8388608)
#define OFF_PERP ((size_t)8388609)
#define OFF_ENC  ((size_t)8388610)
#define OFF_IDX  ((size_t)142606338)

// ---------------- init: zero counts, precompute ||e_k||^2 ----------------
__global__ __launch_bounds__(1024) void vq_init(const float* __restrict__ emb,
                                                int* __restrict__ counts,
                                                float* __restrict__ enorm) {
    int k = threadIdx.x;            // 1024 threads, one per codeword
    counts[k] = 0;
    float s = 0.0f;
    #pragma unroll 8
    for (int d = 0; d < DDIM; ++d) {
        float e = emb[(size_t)d * KCODES + k];   // coalesced across k
        s += e * e;
    }
    enorm[k] = s;
}

// ---------------- main: WMMA distances + argmin + all big outputs --------
__global__ __launch_bounds__(BLOCK) void vq_main(const float* __restrict__ x,
                                                 const float* __restrict__ emb,
                                                 float* __restrict__ out,
                                                 int* __restrict__ counts,
                                                 const float* __restrict__ enorm,
                                                 float* __restrict__ partials) {
    const int tid  = threadIdx.x;
    const int wave = tid >> 5;
    const int lane = tid & 31;
    const int half = lane >> 4;
    const int lrow = lane & 15;
    const int m0   = blockIdx.x * 16;     // this block's 16 rows of x

    __shared__ float s_v[WAVES_PB][16];
    __shared__ int   s_i[WAVES_PB][16];
    __shared__ int   s_idx[16];
    __shared__ float s_ls[WAVES_PB];

    // ---- load A fragments once: 16 chunks of K=4 over D=64 ----
    // ISA layout (32-bit A 16x4): lane(row)=lrow; VGPR0 holds K=2*half, VGPR1 K=2*half+1
    const v2f* xr2 = (const v2f*)(x + (size_t)(m0 + lrow) * DDIM);
    v2f a[16];
    #pragma unroll
    for (int dd = 0; dd < 16; ++dd)
        a[dd] = xr2[dd * 2 + half];

    float minv[8];
    int   mini[8];
    #pragma unroll
    for (int r = 0; r < 8; ++r) { minv[r] = 3.4e38f; mini[r] = 0; }

    // ---- scan this wave's 8 column tiles of 16 codewords ----
    for (int t = 0; t < TILES_PW; ++t) {
        const int n0 = (wave * TILES_PW + t) * 16;
        v8f c = {0.f, 0.f, 0.f, 0.f, 0.f, 0.f, 0.f, 0.f};
        #pragma unroll
        for (int dd = 0; dd < 16; ++dd) {
            const int d = dd * 4 + 2 * half;
            v2f b;   // B 4x16: lane(col)=lrow; VGPR0 K=2*half, VGPR1 K=2*half+1
            b.x = emb[(size_t)(d    ) * KCODES + n0 + lrow];
            b.y = emb[(size_t)(d + 1) * KCODES + n0 + lrow];
            c = __builtin_amdgcn_wmma_f32_16x16x4_f32(
                    false, a[dd], false, b, (short)0, c, false, false);
        }
        const float en = enorm[n0 + lrow];
        #pragma unroll
        for (int r = 0; r < 8; ++r) {
            // dist (up to +||x||^2 const) = ||e||^2 - 2 x.e ; first-min tiebreak
            float s = en - 2.0f * c[r];      // C VGPR r <-> row m0+r+8*half, col n0+lrow
            if (s < minv[r]) { minv[r] = s; mini[r] = n0 + lrow; }
        }
    }

    // ---- butterfly reduce across the 16 lanes sharing each row ----
    #pragma unroll
    for (int off = 8; off >= 1; off >>= 1) {
        #pragma unroll
        for (int r = 0; r < 8; ++r) {
            float ov = __shfl_xor(minv[r], off, 32);
            int   oi = __shfl_xor(mini[r], off, 32);
            if (ov < minv[r] || (ov == minv[r] && oi < mini[r])) {
                minv[r] = ov; mini[r] = oi;
            }
        }
    }
    if (lrow == 0) {
        #pragma unroll
        for (int r = 0; r < 8; ++r) {
            s_v[wave][r + 8 * half] = minv[r];
            s_i[wave][r + 8 * half] = mini[r];
        }
    }
    __syncthreads();

    // ---- combine across waves; emit indices + counts ----
    if (tid < 16) {
        float bv = s_v[0][tid];
        int   bi = s_i[0][tid];
        #pragma unroll
        for (int w = 1; w < WAVES_PB; ++w) {
            float v = s_v[w][tid]; int i = s_i[w][tid];
            if (v < bv || (v == bv && i < bi)) { bv = v; bi = i; }
        }
        s_idx[tid] = bi;
        out[OFF_IDX + m0 + tid] = (float)bi;     // indices as float payload
        atomicAdd(&counts[bi], 1);               // integer atomic: deterministic
    }
    __syncthreads();

    // ---- stream one-hot rows (536 MB total): nontemporal v2f stores ----
    {
        v2f* enc2 = (v2f*)(out + OFF_ENC + (size_t)m0 * KCODES); // 8B aligned
        #pragma unroll 4
        for (int e2 = tid; e2 < (16 * KCODES) / 2; e2 += BLOCK) {
            const int e   = e2 << 1;
            const int row = e >> 10;
            const int k   = e & (KCODES - 1);
            const int idx = s_idx[row];
            v2f v;
            v.x = (k     == idx) ? 1.0f : 0.0f;
            v.y = (k + 1 == idx) ? 1.0f : 0.0f;
            __builtin_nontemporal_store(v, enc2 + e2);
        }
    }

    // ---- quantized gather + commitment-loss partial ----
    float lsum = 0.0f;
    #pragma unroll
    for (int it = 0; it < (16 * DDIM) / BLOCK; ++it) {
        const int e   = tid + it * BLOCK;
        const int row = e >> 6;
        const int d   = e & (DDIM - 1);
        const int idx = s_idx[row];
        const float q  = emb[(size_t)d * KCODES + idx];
        const float xv = x[(size_t)(m0 + row) * DDIM + d];
        const float df = q - xv;
        lsum += df * df;
        __builtin_nontemporal_store(q, out + OFF_Q + (size_t)(m0 + row) * DDIM + d);
    }
    #pragma unroll
    for (int off = 16; off >= 1; off >>= 1)
        lsum += __shfl_xor(lsum, off, 32);
    if (lane == 0) s_ls[wave] = lsum;
    __syncthreads();
    if (tid == 0) {
        float t = 0.0f;
        #pragma unroll
        for (int w = 0; w < WAVES_PB; ++w) t += s_ls[w];
        partials[blockIdx.x] = t;
    }
}

// ---------------- finalize: loss mean + perplexity ----------------------
__global__ __launch_bounds__(1024) void vq_finalize(const int* __restrict__ counts,
                                                    const float* __restrict__ partials,
                                                    float* __restrict__ out) {
    __shared__ float red[1024];
    const int tid = threadIdx.x;

    float s = 0.0f;
    for (int i = tid; i < NBLOCKS; i += 1024) s += partials[i];
    red[tid] = s;
    __syncthreads();
    for (int st = 512; st > 0; st >>= 1) {
        if (tid < st) red[tid] += red[tid + st];
        __syncthreads();
    }
    if (tid == 0)
        out[OFF_LOSS] = 0.25f * red[0] / (float)((size_t)N_ROWS * DDIM);
    __syncthreads();

    const float p = (float)counts[tid] * (1.0f / (float)N_ROWS);
    red[tid] = p * __logf(p + 1e-10f);
    __syncthreads();
    for (int st = 512; st > 0; st >>= 1) {
        if (tid < st) red[tid] += red[tid + st];
        __syncthreads();
    }
    if (tid == 0)
        out[OFF_PERP] = __expf(-red[0]);
}

extern "C" void kernel_launch(void* const* d_in, const int* in_sizes, int n_in,
                              void* d_out, int out_size, void* d_ws, size_t ws_size,
                              hipStream_t stream) {
    const float* x   = (const float*)d_in[0];   // [32,64,64,64] fp32
    const float* emb = (const float*)d_in[1];   // [64,1024] fp32
    float* out = (float*)d_out;

    int*   counts   = (int*)d_ws;                              // 1024 ints
    float* enorm    = (float*)((char*)d_ws + 4096);            // 1024 floats
    float* partials = (float*)((char*)d_ws + 8192);            // 8192 floats

    vq_init<<<1, 1024, 0, stream>>>(emb, counts, enorm);
    vq_main<<<NBLOCKS, BLOCK, 0, stream>>>(x, emb, out, counts, enorm, partials);
    vq_finalize<<<1, 1024, 0, stream>>>(counts, partials, out);
}